// VNN_cell_PLUS_28913719836808
// MI455X (gfx1250) — compile-verified
//
#include <hip/hip_runtime.h>
#include <math.h>

typedef __attribute__((ext_vector_type(2))) float v2f;
typedef __attribute__((ext_vector_type(8))) float v8f;

#define NGENES 3000
#define BATCH  2048
#define GFD    3
#define DDRUG  128
#define NG1    256
#define KC1    32
#define NH1    10
#define NG2    64
#define KC2    8
#define NH2    10
#define NOUT   51
#define KD1    (KC1*GFD)   // 96
#define KD2    (KC2*NH1)   // 80
#define KD3    (NG2*NH2)   // 640

// mish(x) = x * tanh(softplus(x)); with t = e^x:
// tanh(log1p(t)) = t(t+2) / (t(t+2)+2)  -> one v_exp + one v_rcp
__device__ __forceinline__ float mishf(float x) {
    float t  = __expf(x);
    float nm = t * (t + 2.0f);
    float r  = nm * __builtin_amdgcn_rcpf(nm + 2.0f);
    return (x > 20.0f) ? x : x * r;
}

__device__ __forceinline__ float sigmf(float x) {
    return __builtin_amdgcn_rcpf(1.0f + __expf(-x));
}

// ---- Q = drug @ Wq + bq : [B, GF] -----------------------------------------
__global__ __launch_bounds__(256) void qproj_kernel(
    const float* __restrict__ drug, const float* __restrict__ Wq,
    const float* __restrict__ bq, float* __restrict__ Q) {
    int idx = blockIdx.x * 256 + threadIdx.x;      // 0 .. B*GF-1
    int b = idx / GFD, o = idx - b * GFD;
    float acc = bq[o];
    const float* d = drug + b * DDRUG;
    #pragma unroll 4
    for (int i = 0; i < DDRUG; ++i) acc += d[i] * Wq[i * GFD + o];
    Q[idx] = acc;
}

// ---- gf = sigmoid(<Q,K>) * V + features : streaming [N,B,GF] --------------
// 4 (gene,batch) pairs per thread -> b128 loads/stores.
__global__ __launch_bounds__(256) void attn_kernel(
    const float4* __restrict__ feat4, const float4* __restrict__ Q4,
    const float* __restrict__ Wk, const float* __restrict__ bk,
    const float* __restrict__ Wv, const float* __restrict__ bv,
    float4* __restrict__ gf4) {
    int idx = blockIdx.x * 256 + threadIdx.x;      // handles pairs idx*4..idx*4+3
    float wk[9], wv[9], bkv[3], bvv[3];
    #pragma unroll
    for (int i = 0; i < 9; ++i) { wk[i] = Wk[i]; wv[i] = Wv[i]; }
    #pragma unroll
    for (int i = 0; i < 3; ++i) { bkv[i] = bk[i]; bvv[i] = bv[i]; }

    float4 fa = feat4[(size_t)idx * 3 + 0];
    float4 fb = feat4[(size_t)idx * 3 + 1];
    float4 fc = feat4[(size_t)idx * 3 + 2];
    int qb = (idx % (BATCH / 4)) * 3;              // Q float4 base for b0 = 4*(idx%512)
    float4 qa = Q4[qb + 0];
    float4 qb4 = Q4[qb + 1];
    float4 qc = Q4[qb + 2];

    float f[12] = { fa.x, fa.y, fa.z, fa.w, fb.x, fb.y, fb.z, fb.w, fc.x, fc.y, fc.z, fc.w };
    float q[12] = { qa.x, qa.y, qa.z, qa.w, qb4.x, qb4.y, qb4.z, qb4.w, qc.x, qc.y, qc.z, qc.w };
    float o[12];
    #pragma unroll
    for (int p = 0; p < 4; ++p) {
        float f0 = f[p * 3 + 0], f1 = f[p * 3 + 1], f2 = f[p * 3 + 2];
        float k0 = f0 * wk[0] + f1 * wk[3] + f2 * wk[6] + bkv[0];
        float k1 = f0 * wk[1] + f1 * wk[4] + f2 * wk[7] + bkv[1];
        float k2 = f0 * wk[2] + f1 * wk[5] + f2 * wk[8] + bkv[2];
        float v0 = f0 * wv[0] + f1 * wv[3] + f2 * wv[6] + bvv[0];
        float v1 = f0 * wv[1] + f1 * wv[4] + f2 * wv[7] + bvv[1];
        float v2 = f0 * wv[2] + f1 * wv[5] + f2 * wv[8] + bvv[2];
        float s  = q[p * 3 + 0] * k0 + q[p * 3 + 1] * k1 + q[p * 3 + 2] * k2;
        float att = sigmf(s);
        o[p * 3 + 0] = v0 * att + f0;
        o[p * 3 + 1] = v1 * att + f1;
        o[p * 3 + 2] = v2 * att + f2;
    }
    gf4[(size_t)idx * 3 + 0] = make_float4(o[0], o[1], o[2],  o[3]);
    gf4[(size_t)idx * 3 + 1] = make_float4(o[4], o[5], o[6],  o[7]);
    gf4[(size_t)idx * 3 + 2] = make_float4(o[8], o[9], o[10], o[11]);
}

// ---- level-1 group GEMM: per group [2048 x 96] @ [96 x 10], Mish ----------
__global__ __launch_bounds__(256) void lvl1_kernel(
    const float* __restrict__ gf, const float* __restrict__ W1,
    const float* __restrict__ b1, const int* __restrict__ gidx,
    float* __restrict__ h1) {
    __shared__ float sW[KD1 * 16];
    __shared__ float sb[16];
    __shared__ int   cb[KD1];
    int g = blockIdx.x, tid = threadIdx.x;
    for (int i = tid; i < KD1 * 16; i += 256) {
        int kk = i >> 4, n = i & 15;
        sW[i] = (n < NH1) ? W1[(g * KD1 + kk) * NH1 + n] : 0.0f;
    }
    if (tid < 16) sb[tid] = (tid < NH1) ? b1[g * NH1 + tid] : 0.0f;
    if (tid < KD1) {
        int gi = gidx[g * KC1 + tid / GFD];
        cb[tid] = gi * (BATCH * GFD) + tid % GFD;
    }
    __syncthreads();

    int wave = tid >> 5, lane = tid & 31;
    int half = lane >> 4, lr = lane & 15;
    for (int mt = wave; mt < BATCH / 16; mt += 8) {   // uniform per wave
        int m = mt * 16 + lr;                          // A row = lane%16
        v8f acc = {};
        #pragma unroll 4
        for (int s = 0; s < KD1 / 4; ++s) {
            int kA = s * 4 + half * 2;                 // K split over lane halves
            v2f a, bf;
            a.x  = gf[cb[kA]     + m * GFD];
            a.y  = gf[cb[kA + 1] + m * GFD];
            bf.x = sW[kA * 16 + lr];
            bf.y = sW[(kA + 1) * 16 + lr];
            acc = __builtin_amdgcn_wmma_f32_16x16x4_f32(
                false, a, false, bf, (short)0, acc, false, false);
        }
        if (lr < NH1) {                                // column n = lane%16
            int mbase = mt * 16 + half * 8;
            #pragma unroll
            for (int r = 0; r < 8; ++r) {
                float x = acc[r] + sb[lr];
                h1[(g * BATCH + mbase + r) * NH1 + lr] = mishf(x);
            }
        }
    }
}

// ---- level-2 group GEMM: per group [2048 x 80] @ [80 x 10], Mish ----------
__global__ __launch_bounds__(256) void lvl2_kernel(
    const float* __restrict__ h1, const float* __restrict__ W2,
    const float* __restrict__ b2, const int* __restrict__ child,
    float* __restrict__ h2) {
    __shared__ float sW[KD2 * 16];
    __shared__ float sb[16];
    __shared__ int   cb[KD2];
    int g = blockIdx.x, tid = threadIdx.x;
    for (int i = tid; i < KD2 * 16; i += 256) {
        int kk = i >> 4, n = i & 15;
        sW[i] = (n < NH2) ? W2[(g * KD2 + kk) * NH2 + n] : 0.0f;
    }
    if (tid < 16) sb[tid] = (tid < NH2) ? b2[g * NH2 + tid] : 0.0f;
    if (tid < KD2) {
        int c = child[g * KC2 + tid / NH1];
        cb[tid] = c * (BATCH * NH1) + tid % NH1;
    }
    __syncthreads();

    int wave = tid >> 5, lane = tid & 31;
    int half = lane >> 4, lr = lane & 15;
    for (int mt = wave; mt < BATCH / 16; mt += 8) {
        int m = mt * 16 + lr;
        v8f acc = {};
        #pragma unroll 4
        for (int s = 0; s < KD2 / 4; ++s) {
            int kA = s * 4 + half * 2;
            v2f a, bf;
            a.x  = h1[cb[kA]     + m * NH1];
            a.y  = h1[cb[kA + 1] + m * NH1];
            bf.x = sW[kA * 16 + lr];
            bf.y = sW[(kA + 1) * 16 + lr];
            acc = __builtin_amdgcn_wmma_f32_16x16x4_f32(
                false, a, false, bf, (short)0, acc, false, false);
        }
        if (lr < NH2) {
            int mbase = mt * 16 + half * 8;
            #pragma unroll
            for (int r = 0; r < 8; ++r) {
                float x = acc[r] + sb[lr];
                h2[(g * BATCH + mbase + r) * NH2 + lr] = mishf(x);
            }
        }
    }
}

// ---- root GEMM: [2048 x 640] @ [640 x 51], Mish ---------------------------
__global__ __launch_bounds__(32) void root_kernel(
    const float* __restrict__ h2, const float* __restrict__ W3,
    const float* __restrict__ b3, float* __restrict__ h3) {
    int mt = blockIdx.x;                 // 0..127 (batch tiles of 16)
    int nt = blockIdx.y;                 // 0..3   (output tiles of 16)
    int lane = threadIdx.x;
    int half = lane >> 4, lr = lane & 15;
    int n = nt * 16 + lr;
    int m = mt * 16 + lr;
    bool nvalid = (n < NOUT);
    v8f acc = {};
    for (int s = 0; s < KD3 / 4; ++s) {
        int kA = s * 4 + half * 2;
        int c0 = kA / NH2,       f0c = kA - c0 * NH2;
        int c1 = (kA + 1) / NH2, f1c = (kA + 1) - c1 * NH2;
        v2f a, bf;
        a.x  = h2[(c0 * BATCH + m) * NH2 + f0c];
        a.y  = h2[(c1 * BATCH + m) * NH2 + f1c];
        bf.x = nvalid ? W3[kA * NOUT + n]       : 0.0f;
        bf.y = nvalid ? W3[(kA + 1) * NOUT + n] : 0.0f;
        acc = __builtin_amdgcn_wmma_f32_16x16x4_f32(
            false, a, false, bf, (short)0, acc, false, false);
    }
    if (nvalid) {
        int mbase = mt * 16 + half * 8;
        #pragma unroll
        for (int r = 0; r < 8; ++r) {
            float x = acc[r] + b3[n];
            h3[(mbase + r) * NOUT + n] = mishf(x);
        }
    }
}

// ---- training-mode BatchNorm over batch axis, in-place --------------------
// layout: element(b) = g*groupStride + b*H + h,  bid = g*H + h
__global__ __launch_bounds__(256) void bn_kernel(
    float* __restrict__ x, float* __restrict__ out,
    const float* __restrict__ gamma, const float* __restrict__ beta,
    int H, long groupStride) {
    __shared__ float ssum[256], ssq[256];
    int bid = blockIdx.x, tid = threadIdx.x;
    int g = bid / H, h = bid - g * H;
    long base = (long)g * groupStride + h;
    float s = 0.0f, q = 0.0f;
    for (int b = tid; b < BATCH; b += 256) {
        float v = x[base + (long)b * H];
        s += v; q += v * v;
    }
    ssum[tid] = s; ssq[tid] = q;
    __syncthreads();
    for (int off = 128; off > 0; off >>= 1) {
        if (tid < off) { ssum[tid] += ssum[tid + off]; ssq[tid] += ssq[tid + off]; }
        __syncthreads();
    }
    float mean = ssum[0] * (1.0f / BATCH);
    float var  = ssq[0]  * (1.0f / BATCH) - mean * mean;
    float scale = gamma[bid] * rsqrtf(var + 1e-5f);
    float shift = beta[bid] - mean * scale;
    for (int b = tid; b < BATCH; b += 256) {
        long i = base + (long)b * H;
        out[i] = x[i] * scale + shift;
    }
}

extern "C" void kernel_launch(void* const* d_in, const int* in_sizes, int n_in,
                              void* d_out, int out_size, void* d_ws, size_t ws_size,
                              hipStream_t stream) {
    const float* features = (const float*)d_in[0];
    const float* drug     = (const float*)d_in[1];
    const float* Wq = (const float*)d_in[2];
    const float* bq = (const float*)d_in[3];
    const float* Wk = (const float*)d_in[4];
    const float* bk = (const float*)d_in[5];
    const float* Wv = (const float*)d_in[6];
    const float* bv = (const float*)d_in[7];
    const float* W1 = (const float*)d_in[8];
    const float* b1 = (const float*)d_in[9];
    const float* g1 = (const float*)d_in[10];
    const float* be1 = (const float*)d_in[11];
    const float* W2 = (const float*)d_in[12];
    const float* b2 = (const float*)d_in[13];
    const float* g2 = (const float*)d_in[14];
    const float* be2 = (const float*)d_in[15];
    const float* W3 = (const float*)d_in[16];
    const float* b3 = (const float*)d_in[17];
    const float* g3 = (const float*)d_in[18];
    const float* be3 = (const float*)d_in[19];
    const int* gene_idx = (const int*)d_in[20];
    const int* child2   = (const int*)d_in[21];

    float* ws = (float*)d_ws;
    float* Q  = ws;                                      // 6144 floats
    float* gf = ws + 8192;                               // N*B*GF = 18,432,000
    float* h1 = gf + (size_t)NGENES * BATCH * GFD;       // G1*B*H1 = 5,242,880
    float* h2 = h1 + (size_t)NG1 * BATCH * NH1;          // G2*B*H2 = 1,310,720
    float* out = (float*)d_out;                          // [B, 51]

    qproj_kernel<<<(BATCH * GFD) / 256, 256, 0, stream>>>(drug, Wq, bq, Q);
    attn_kernel<<<(NGENES * BATCH / 4) / 256, 256, 0, stream>>>(
        (const float4*)features, (const float4*)Q, Wk, bk, Wv, bv, (float4*)gf);
    lvl1_kernel<<<NG1, 256, 0, stream>>>(gf, W1, b1, gene_idx, h1);
    bn_kernel<<<NG1 * NH1, 256, 0, stream>>>(h1, h1, g1, be1, NH1, (long)BATCH * NH1);
    lvl2_kernel<<<NG2, 256, 0, stream>>>(h1, W2, b2, child2, h2);
    bn_kernel<<<NG2 * NH2, 256, 0, stream>>>(h2, h2, g2, be2, NH2, (long)BATCH * NH2);
    root_kernel<<<dim3(BATCH / 16, 4), 32, 0, stream>>>(h2, W3, b3, out);
    bn_kernel<<<NOUT, 256, 0, stream>>>(out, out, g3, be3, NOUT, 0L);
}